// GraphAttentionLayer_71073118814527
// MI455X (gfx1250) — compile-verified
//
#include <hip/hip_runtime.h>

#define GN 8192
#define GIN_F 512
#define GOUT_F 64

typedef __attribute__((ext_vector_type(2))) float v2f;
typedef __attribute__((ext_vector_type(8))) float v8f;

__device__ __forceinline__ v8f wmma4(v2f a, v2f b, v8f c) {
  // D = A(16x4 f32) * B(4x16 f32) + C(16x16 f32)
  return __builtin_amdgcn_wmma_f32_16x16x4_f32(false, a, false, b, (short)0, c,
                                               false, false);
}

// ---------------------------------------------------------------------------
// Kernel 1: WH = h @ W   [8192x512]x[512x64] via V_WMMA_F32_16X16X4_F32
// 64 blocks x 256 threads; each wave computes a 16x64 strip.
// ---------------------------------------------------------------------------
__global__ void __launch_bounds__(256) k_wh(const float* __restrict__ h,
                                            const float* __restrict__ W,
                                            float* __restrict__ WH) {
  const int lane = threadIdx.x & 31;
  const int wave = threadIdx.x >> 5;
  const int hi = lane >> 4;   // 0: K={0,1}, 1: K={2,3}
  const int c16 = lane & 15;
  const int arow = blockIdx.x * 128 + wave * 16 + c16;  // A-matrix row for lane

  v8f acc[4] = {};
  for (int k = 0; k < GIN_F; k += 4) {
    v2f a;
    const float* hp = h + (size_t)arow * GIN_F + k + 2 * hi;
    a.x = hp[0];
    a.y = hp[1];
#pragma unroll
    for (int nt = 0; nt < 4; ++nt) {
      const int col = nt * 16 + c16;
      v2f b;
      b.x = W[(size_t)(k + 2 * hi) * GOUT_F + col];
      b.y = W[(size_t)(k + 2 * hi + 1) * GOUT_F + col];
      acc[nt] = wmma4(a, b, acc[nt]);
    }
  }
  // C layout: VGPR r -> row (r + 8*hi), col c16
  const int rbase = blockIdx.x * 128 + wave * 16 + 8 * hi;
#pragma unroll
  for (int nt = 0; nt < 4; ++nt)
#pragma unroll
    for (int r = 0; r < 8; ++r)
      WH[(size_t)(rbase + r) * GOUT_F + nt * 16 + c16] = acc[nt][r];
}

// ---------------------------------------------------------------------------
// Kernel 2: wh1 = WH @ a[:64], wh2 = WH @ a[64:]. One wave per row.
// ---------------------------------------------------------------------------
__global__ void __launch_bounds__(256) k_attvec(const float* __restrict__ WH,
                                                const float* __restrict__ a,
                                                float* __restrict__ wh1,
                                                float* __restrict__ wh2) {
  const int lane = threadIdx.x & 31;
  const int wave = threadIdx.x >> 5;
  const int row = blockIdx.x * 8 + wave;
  const float v0 = WH[(size_t)row * GOUT_F + lane];
  const float v1 = WH[(size_t)row * GOUT_F + lane + 32];
  float p1 = v0 * a[lane] + v1 * a[lane + 32];
  float p2 = v0 * a[GOUT_F + lane] + v1 * a[GOUT_F + lane + 32];
#pragma unroll
  for (int off = 16; off > 0; off >>= 1) {
    p1 += __shfl_xor(p1, off, 32);
    p2 += __shfl_xor(p2, off, 32);
  }
  if (lane == 0) {
    wh1[row] = p1;
    wh2[row] = p2;
  }
}

// ---------------------------------------------------------------------------
// Kernel 3: per-row softmax stats + adjacency bitmask (single adjacency read).
// Each block handles one row i: writes maskbits[i][0..255] (256 dwords = 8192
// bits), and cm[i] = m_i + log(sum_j exp(val_ij - m_i)).
// ---------------------------------------------------------------------------
__global__ void __launch_bounds__(256) k_rowstats(const int* __restrict__ adj,
                                                  const float* __restrict__ wh1,
                                                  const float* __restrict__ wh2,
                                                  unsigned* __restrict__ maskbits,
                                                  float* __restrict__ cmrow) {
  __shared__ float red[256];
  __shared__ unsigned lmask[256];
  const int i = blockIdx.x;
  const int tid = threadIdx.x;
  const int lane = tid & 31;
  const int wave = tid >> 5;
  const float w1 = wh1[i];
  const int* arow = adj + (size_t)i * GN;

  float mloc = -3.0e38f;
  for (int it = 0; it < 32; ++it) {
    const int j = it * 256 + tid;
    const bool on = arow[j] > 0;
    const float x = w1 + wh2[j];
    const float e = x > 0.f ? x : 0.01f * x;  // leaky_relu(0.01)
    mloc = fmaxf(mloc, on ? e : -1.0e9f);
    const unsigned mw = __builtin_amdgcn_ballot_w32(on);
    if (lane == 0) lmask[it * 8 + wave] = mw;
  }
  red[tid] = mloc;
  __syncthreads();
  for (int off = 128; off > 0; off >>= 1) {
    if (tid < off) red[tid] = fmaxf(red[tid], red[tid + off]);
    __syncthreads();
  }
  const float m = red[0];
  __syncthreads();

  float sloc = 0.f;
  for (int it = 0; it < 32; ++it) {
    const int j = it * 256 + tid;
    const unsigned mw = lmask[it * 8 + wave];
    const float x = w1 + wh2[j];
    const float e = x > 0.f ? x : 0.01f * x;
    const float v = ((mw >> lane) & 1u) ? e : -1.0e9f;
    sloc += __expf(v - m);  // exp(-1e9 - m) flushes to 0 like the reference
  }
  red[tid] = sloc;
  __syncthreads();
  for (int off = 128; off > 0; off >>= 1) {
    if (tid < off) red[tid] += red[tid + off];
    __syncthreads();
  }
  if (tid == 0) cmrow[i] = m + __logf(red[0]);  // p = exp(e - cm)
  maskbits[(size_t)i * 256 + tid] = lmask[tid];
}

// ---------------------------------------------------------------------------
// Kernel 4: out = elu( softmax(P) @ WH ), P generated on the fly from bitmask.
// 128 blocks x 256 threads. Block owns 64 rows; wave w covers K in
// [w*1024, (w+1)*1024); per wave: 4 M-tiles x 4 N-tiles of f32 WMMA C.
// Cross-wave reduction via LDS ds_add_f32, then fused ELU + store.
// ---------------------------------------------------------------------------
__global__ void __launch_bounds__(256) k_out(const float* __restrict__ WH,
                                             const float* __restrict__ wh1,
                                             const float* __restrict__ wh2,
                                             const float* __restrict__ cmrow,
                                             const unsigned* __restrict__ maskbits,
                                             float* __restrict__ out) {
  __shared__ float cbuf[64 * 64];
  const int tid = threadIdx.x;
  const int lane = tid & 31;
  const int wave = tid >> 5;
  const int hi = lane >> 4;
  const int c16 = lane & 15;
  const int R0 = blockIdx.x * 64;

  int rowm[4];
  float w1m[4], cmm[4];
#pragma unroll
  for (int mt = 0; mt < 4; ++mt) {
    const int r = R0 + mt * 16 + c16;
    rowm[mt] = r;
    w1m[mt] = wh1[r];
    cmm[mt] = cmrow[r];
  }

  v8f acc[4][4] = {};
  const int kbase = wave * 1024;
  for (int kb = 0; kb < 32; ++kb) {
    const int k32 = kbase + kb * 32;
    unsigned mw[4];
#pragma unroll
    for (int mt = 0; mt < 4; ++mt)
      mw[mt] = maskbits[(size_t)rowm[mt] * 256 + (k32 >> 5)];
#pragma unroll
    for (int t = 0; t < 8; ++t) {
      const int j0 = k32 + t * 4 + 2 * hi;  // this lane's two K columns
      const float w2a = wh2[j0];
      const float w2b = wh2[j0 + 1];
      v2f B[4];
#pragma unroll
      for (int nt = 0; nt < 4; ++nt) {
        const int col = nt * 16 + c16;
        B[nt].x = WH[(size_t)j0 * GOUT_F + col];
        B[nt].y = WH[(size_t)(j0 + 1) * GOUT_F + col];
      }
      const int sh = t * 4 + 2 * hi;
#pragma unroll
      for (int mt = 0; mt < 4; ++mt) {
        const float x0 = w1m[mt] + w2a;
        const float x1 = w1m[mt] + w2b;
        const float e0 = x0 > 0.f ? x0 : 0.01f * x0;
        const float e1 = x1 > 0.f ? x1 : 0.01f * x1;
        v2f A;
        A.x = ((mw[mt] >> sh) & 1u) ? __expf(e0 - cmm[mt]) : 0.f;
        A.y = ((mw[mt] >> (sh + 1)) & 1u) ? __expf(e1 - cmm[mt]) : 0.f;
#pragma unroll
        for (int nt = 0; nt < 4; ++nt) acc[mt][nt] = wmma4(A, B[nt], acc[mt][nt]);
      }
    }
  }

  for (int idx = tid; idx < 4096; idx += 256) cbuf[idx] = 0.f;
  __syncthreads();
#pragma unroll
  for (int mt = 0; mt < 4; ++mt)
#pragma unroll
    for (int nt = 0; nt < 4; ++nt)
#pragma unroll
      for (int r = 0; r < 8; ++r)
        atomicAdd(&cbuf[(mt * 16 + r + 8 * hi) * 64 + nt * 16 + c16],
                  acc[mt][nt][r]);
  __syncthreads();
  for (int idx = tid; idx < 4096; idx += 256) {
    const float v = cbuf[idx];
    out[(size_t)(R0 + (idx >> 6)) * GOUT_F + (idx & 63)] =
        v > 0.f ? v : __expf(v) - 1.f;  // ELU(alpha=1)
  }
}

// ---------------------------------------------------------------------------
extern "C" void kernel_launch(void* const* d_in, const int* in_sizes, int n_in,
                              void* d_out, int out_size, void* d_ws,
                              size_t ws_size, hipStream_t stream) {
  const float* h = (const float*)d_in[0];     // [8192, 512] f32
  const int* adj = (const int*)d_in[1];       // [8192, 8192] i32
  const float* W = (const float*)d_in[2];     // [512, 64] f32
  const float* a = (const float*)d_in[3];     // [128, 1] f32
  float* out = (float*)d_out;                 // [8192, 64] f32

  // Workspace layout (~10.2 MB)
  float* WH = (float*)d_ws;                   // 8192*64 f32 = 2 MB
  float* wh1 = WH + (size_t)GN * GOUT_F;      // 8192 f32
  float* wh2 = wh1 + GN;                      // 8192 f32
  float* cm = wh2 + GN;                       // 8192 f32
  unsigned* maskb = (unsigned*)(cm + GN);     // 8192*256 u32 = 8 MB

  k_wh<<<64, 256, 0, stream>>>(h, W, WH);
  k_attvec<<<1024, 256, 0, stream>>>(WH, a, wh1, wh2);
  k_rowstats<<<GN, 256, 0, stream>>>(adj, wh1, wh2, maskb, cm);
  k_out<<<128, 256, 0, stream>>>(WH, wh1, wh2, cm, maskb, out);
}